// NSLoss_79207786873168
// MI455X (gfx1250) — compile-verified
//
#include <hip/hip_runtime.h>

// NSLoss (negative-sampling loss) for MI455X / gfx1250, wave32.
//
// Gather-bandwidth/latency bound: ~233 MB of row gathers @ 23.3 TB/s (~10 us),
// only 0.1 GFLOP useful. Dot products computed as diagonals of 16x16 tiles via
// V_WMMA_F32_16X16X4_F32 (full f32 precision; wasted MACs are free at
// AI = 0.45 FLOP/B). Source-level software pipelining (load step s+1 before
// WMMAs of step s) forces partial s_wait_loadcnt instead of the serial
// load -> s_wait_loadcnt 0 -> wmma pattern, keeping 7-14 gathers in flight
// per wave.

typedef float v2f __attribute__((ext_vector_type(2)));
typedef float v8f __attribute__((ext_vector_type(8)));

#define EMB 128
#define NUM_SAMPLED 5
#define ROWS_PER_TILE 16
#define WAVES_PER_BLOCK 8
#define KSTEPS (EMB / 4)  // 32 k-steps of 4

__device__ __forceinline__ float log_sigmoid(float x) {
    // log(sigmoid(x)) = min(x,0) - log1p(exp(-|x|))  (numerically stable)
    return fminf(x, 0.0f) - log1pf(__expf(-fabsf(x)));
}

// 16x16 f32 C/D layout (wave32): lanes 0-15 hold N=lane with M=vgpr,
// lanes 16-31 hold N=lane-16 with M=vgpr+8.  Diagonal (m,m) lives in
// lanes 0-7 (vgpr=lane) and lanes 24-31 (vgpr=lane-24).
__device__ __forceinline__ float diag_extract(v8f c, int lane) {
    int v = (lane < 8) ? lane : (lane - 24);
    float d = c[0];
    d = (v == 1) ? c[1] : d;
    d = (v == 2) ? c[2] : d;
    d = (v == 3) ? c[3] : d;
    d = (v == 4) ? c[4] : d;
    d = (v == 5) ? c[5] : d;
    d = (v == 6) ? c[6] : d;
    d = (v == 7) ? c[7] : d;
    return d;
}

__device__ __forceinline__ v2f ld2(const float* p) {
    return *(const v2f*)p;
}

__global__ void __launch_bounds__(256)
nsloss_tile_kernel(const float* __restrict__ embs,
                   const int*   __restrict__ label,
                   const int*   __restrict__ negs,
                   const float* __restrict__ W,
                   float*       __restrict__ partial) {
    const int lane = threadIdx.x & 31;
    const int wave = threadIdx.x >> 5;
    const int tile = blockIdx.x * WAVES_PER_BLOCK + wave;
    const int row0 = tile * ROWS_PER_TILE;

    // A-matrix (16x4 f32) layout: lane m (0-15) supplies K = k0,k0+1 of row m,
    // lane m+16 supplies K = k0+2,k0+3 of row m.  B (4x16) mirrors this with
    // N = lane&15.  One aligned 8B load per lane per operand per k-step.
    const int r     = row0 + (lane & 15);
    const int khalf = (lane >> 4) * 2;  // 0 or 2

    const float* pE = embs + (size_t)r * EMB + khalf;
    const float* p0 = W + (size_t)label[r] * EMB + khalf;
    const float* p1 = W + (size_t)negs[(size_t)r * NUM_SAMPLED + 0] * EMB + khalf;
    const float* p2 = W + (size_t)negs[(size_t)r * NUM_SAMPLED + 1] * EMB + khalf;
    const float* p3 = W + (size_t)negs[(size_t)r * NUM_SAMPLED + 2] * EMB + khalf;
    const float* p4 = W + (size_t)negs[(size_t)r * NUM_SAMPLED + 3] * EMB + khalf;
    const float* p5 = W + (size_t)negs[(size_t)r * NUM_SAMPLED + 4] * EMB + khalf;

    v8f c0 = {}, c1 = {}, c2 = {}, c3 = {}, c4 = {}, c5 = {};

    // ---- software pipeline: prologue loads for step 0 ----
    v2f b_c  = ld2(pE);
    v2f a_c0 = ld2(p0);
    v2f a_c1 = ld2(p1);
    v2f a_c2 = ld2(p2);
    v2f a_c3 = ld2(p3);
    v2f a_c4 = ld2(p4);
    v2f a_c5 = ld2(p5);

#pragma unroll
    for (int s = 0; s < KSTEPS - 1; ++s) {
        const int kn = (s + 1) * 4;
        // Issue next step's 7 loads BEFORE consuming the current step's data:
        // the wait before the wmma group is then a partial loadcnt, keeping
        // the next gathers in flight under the WMMAs.
        v2f b_n  = ld2(pE + kn);
        v2f a_n0 = ld2(p0 + kn);
        v2f a_n1 = ld2(p1 + kn);
        v2f a_n2 = ld2(p2 + kn);
        v2f a_n3 = ld2(p3 + kn);
        v2f a_n4 = ld2(p4 + kn);
        v2f a_n5 = ld2(p5 + kn);

        // 6 independent accumulator chains -> no WMMA->WMMA RAW hazards.
        c0 = __builtin_amdgcn_wmma_f32_16x16x4_f32(false, a_c0, false, b_c, (short)0, c0, false, false);
        c1 = __builtin_amdgcn_wmma_f32_16x16x4_f32(false, a_c1, false, b_c, (short)0, c1, false, false);
        c2 = __builtin_amdgcn_wmma_f32_16x16x4_f32(false, a_c2, false, b_c, (short)0, c2, false, false);
        c3 = __builtin_amdgcn_wmma_f32_16x16x4_f32(false, a_c3, false, b_c, (short)0, c3, false, false);
        c4 = __builtin_amdgcn_wmma_f32_16x16x4_f32(false, a_c4, false, b_c, (short)0, c4, false, false);
        c5 = __builtin_amdgcn_wmma_f32_16x16x4_f32(false, a_c5, false, b_c, (short)0, c5, false, false);

        // rotate (free under full unroll / SSA renaming)
        b_c  = b_n;
        a_c0 = a_n0; a_c1 = a_n1; a_c2 = a_n2;
        a_c3 = a_n3; a_c4 = a_n4; a_c5 = a_n5;
    }

    // ---- epilogue: final step's WMMAs ----
    c0 = __builtin_amdgcn_wmma_f32_16x16x4_f32(false, a_c0, false, b_c, (short)0, c0, false, false);
    c1 = __builtin_amdgcn_wmma_f32_16x16x4_f32(false, a_c1, false, b_c, (short)0, c1, false, false);
    c2 = __builtin_amdgcn_wmma_f32_16x16x4_f32(false, a_c2, false, b_c, (short)0, c2, false, false);
    c3 = __builtin_amdgcn_wmma_f32_16x16x4_f32(false, a_c3, false, b_c, (short)0, c3, false, false);
    c4 = __builtin_amdgcn_wmma_f32_16x16x4_f32(false, a_c4, false, b_c, (short)0, c4, false, false);
    c5 = __builtin_amdgcn_wmma_f32_16x16x4_f32(false, a_c5, false, b_c, (short)0, c5, false, false);

    // Per-row loss from the tile diagonals (branch-free; masked after).
    float loss = log_sigmoid(diag_extract(c0, lane));          // positive term
    loss += log_sigmoid(-diag_extract(c1, lane));              // negatives: score = -<W,e>
    loss += log_sigmoid(-diag_extract(c2, lane));
    loss += log_sigmoid(-diag_extract(c3, lane));
    loss += log_sigmoid(-diag_extract(c4, lane));
    loss += log_sigmoid(-diag_extract(c5, lane));

    const bool active = (lane < 8) || (lane >= 24);            // lanes holding a diagonal
    float val = active ? loss : 0.0f;

    // Wave32 reduction (fixed order -> deterministic).
#pragma unroll
    for (int off = 16; off > 0; off >>= 1)
        val += __shfl_xor(val, off, 32);

    if (lane == 0) partial[tile] = val;
}

__global__ void __launch_bounds__(256)
nsloss_reduce_kernel(const float* __restrict__ partial, int n,
                     float* __restrict__ out, float scale) {
    __shared__ float s[256];
    float acc = 0.0f;
    for (int i = threadIdx.x; i < n; i += 256) acc += partial[i];
    s[threadIdx.x] = acc;
    __syncthreads();
#pragma unroll
    for (int stride = 128; stride > 0; stride >>= 1) {
        if ((int)threadIdx.x < stride) s[threadIdx.x] += s[threadIdx.x + stride];
        __syncthreads();
    }
    if (threadIdx.x == 0) out[0] = s[0] * scale;
}

extern "C" void kernel_launch(void* const* d_in, const int* in_sizes, int n_in,
                              void* d_out, int out_size, void* d_ws, size_t ws_size,
                              hipStream_t stream) {
    // setup_inputs order: input (unused by reference), embs, label, negs, W
    const float* embs  = (const float*)d_in[1];
    const int*   label = (const int*)  d_in[2];
    const int*   negs  = (const int*)  d_in[3];
    const float* W     = (const float*)d_in[4];

    const int n     = in_sizes[0];               // 65536 rows
    const int tiles = n / ROWS_PER_TILE;         // 4096
    float* partial  = (float*)d_ws;              // 4096 floats of scratch

    const int blocks = tiles / WAVES_PER_BLOCK;  // 512 blocks x 8 waves, exact
    nsloss_tile_kernel<<<blocks, 256, 0, stream>>>(embs, label, negs, W, partial);
    nsloss_reduce_kernel<<<1, 256, 0, stream>>>(partial, tiles, (float*)d_out,
                                                -1.0f / (float)n);
}